// NRIMLPDecoder_9397388444177
// MI455X (gfx1250) — compile-verified
//
#include <hip/hip_runtime.h>
#include <hip/hip_bf16.h>

// Fused NRI MLP decoder for MI455X (gfx1250, wave32, WMMA).
// One block = one (b, t) pair; grid = (49, 8). 256 threads = 8 waves.
// Edge MLP (8->64->64, fp32 WMMA 16x16x4) + weighted receiver scatter (LDS)
// + node MLP (68->64->64->4, fp32 WMMA) + residual, all in one kernel.
// Avoids materializing all_msgs (413 MB) entirely.

typedef float v2f __attribute__((ext_vector_type(2)));
typedef float v8f __attribute__((ext_vector_type(8)));

#define B_   8
#define N_   64
#define T_   50
#define D_   4
#define H_   64
#define E_   4032   // N*(N-1)
#define NE_  63     // edges per receiver

// ---- LDS layout (float offsets; all even -> 8B aligned for ds b64) ----
#define OFF_XS   0                       // 64 x 4            (256)
#define AGG_STR  68
#define OFF_AGG  256                     // 64 x 68           (4352)
#define W2_STR   68
#define OFF_W2   (OFF_AGG + 64*AGG_STR)  // fc2_w[1] 64x68    (4352) -> 4608
#define O1_STR   70
#define OFF_O1   (OFF_W2 + 64*W2_STR)    // out1_w 64x70      (4480) -> 8960
#define O2_STR   68
#define OFF_O2   (OFF_O1 + 64*O1_STR)    // out2_w 64x68      (4352) -> 13440
#define HB_STR   68
#define OFF_HB   (OFF_O2 + 64*O2_STR)    // 8 waves x 16x68   (8704) -> 17792
#define OFF_WB   (OFF_HB + 8*16*HB_STR)  // 8 x 64 edge wts   (512)  -> 26496
#define OFF_O3W  (OFF_WB + 8*64)         // out3_w 4x64       (256)  -> 27008
#define OFF_O3B  (OFF_O3W + 256)         // 4                        -> 27264
#define OFF_O1B  (OFF_O3B + 4)           // 64                       -> 27268
#define OFF_O2B  (OFF_O1B + 64)          // 64                       -> 27332
#define LDS_FLOATS (OFF_O2B + 64)        // 27396 floats = 109584 B (< 320KB WGP LDS)

__device__ __forceinline__ v8f wmma4(v2f a, v2f b, v8f c) {
  // D = A(16x4,f32) * B(4x16,f32) + C(16x16,f32)
  return __builtin_amdgcn_wmma_f32_16x16x4_f32(
      false, a, false, b, (short)0, c, false, false);
}

__device__ __forceinline__ void sched_fence() { asm volatile("" ::: "memory"); }

__global__ __launch_bounds__(256)
void nri_decoder_kernel(const float* __restrict__ inputs,
                        const float* __restrict__ rel_type,
                        const float* __restrict__ fc1w, const float* __restrict__ fc1b,
                        const float* __restrict__ fc2w, const float* __restrict__ fc2b,
                        const float* __restrict__ o1w,  const float* __restrict__ o1b,
                        const float* __restrict__ o2w,  const float* __restrict__ o2b,
                        const float* __restrict__ o3w,  const float* __restrict__ o3b,
                        float* __restrict__ out)
{
  extern __shared__ float lds[];
  const int t    = blockIdx.x;          // 0..48  (time step producing x_tp1)
  const int b    = blockIdx.y;          // 0..7
  const int tid  = threadIdx.x;         // 0..255
  const int wave = tid >> 5;            // 0..7 (wave32)
  const int lane = tid & 31;
  const int l15  = lane & 15;
  const bool hi  = lane >= 16;
  const int ko   = hi ? 2 : 0;          // K-offset within fp32 A/B fragments

  // ---------------- cooperative LDS staging ----------------
  { // x_in[b,t,n,:] : inputs layout (B,N,T,D)
    int n = tid >> 2, d = tid & 3;
    lds[OFF_XS + tid] = inputs[((b*N_ + n)*T_ + t)*D_ + d];
  }
  for (int i = tid; i < 64*AGG_STR; i += 256) lds[OFF_AGG + i] = 0.f;
  for (int i = tid; i < 64*64; i += 256) {          // fc2_w[1][f][k]
    int f = i >> 6, k = i & 63;
    lds[OFF_W2 + f*W2_STR + k] = fc2w[(H_ + f)*H_ + k];
  }
  for (int i = tid; i < 64*68; i += 256) {          // out1_w[f][0..67]
    int f = i / 68, k = i - f*68;
    lds[OFF_O1 + f*O1_STR + k] = o1w[f*68 + k];
  }
  for (int i = tid; i < 64*64; i += 256) {          // out2_w[f][k]
    int f = i >> 6, k = i & 63;
    lds[OFF_O2 + f*O2_STR + k] = o2w[f*64 + k];
  }
  lds[OFF_O3W + tid < 0 ? 0 : OFF_O3W + tid] = (tid < 256) ? o3w[tid] : 0.f; // out3_w 4x64
  if (tid < 4)  lds[OFF_O3B + tid] = o3b[tid];
  if (tid < 64) lds[OFF_O1B + tid] = o1b[tid];
  if (tid < 64) lds[OFF_O2B + tid] = o2b[tid];
  __syncthreads();

  // ---------------- per-wave constant fragments ----------------
  // fc1 B-fragments: B[d][f] = fc1_w[1][f][d]; lanes<16 hold rows d0,d0+1, lanes>=16 d0+2,d0+3.
  v2f b1f[4][2];
#pragma unroll
  for (int ct = 0; ct < 4; ++ct)
#pragma unroll
    for (int kc = 0; kc < 2; ++kc)
      b1f[ct][kc] = *(const v2f*)(fc1w + (H_ + 16*ct + l15)*8 + 4*kc + ko);
  float b1v[4], b2v[4];
#pragma unroll
  for (int ct = 0; ct < 4; ++ct) {
    b1v[ct] = fc1b[H_ + 16*ct + l15];
    b2v[ct] = fc2b[H_ + 16*ct + l15];
  }

  float* hb = &lds[OFF_HB + wave*16*HB_STR];  // per-wave 16x68 staging tile

  // ---------------- edge phase: 8 receivers per wave ----------------
#pragma unroll 1
  for (int ri = 0; ri < 8; ++ri) {
    const int r = 8*ri + wave;  // receiver node (wave-private agg row)
    { // stage this receiver's 63 rel_type weights (+1 zero pad) -> wave-private wbuf
      int j0 = lane, j1 = lane + 32;
      lds[OFF_WB + wave*64 + j0] =
          (j0 < NE_) ? rel_type[(b*E_ + r*NE_ + j0)*2 + 1] : 0.f;
      lds[OFF_WB + wave*64 + j1] =
          (j1 < NE_) ? rel_type[(b*E_ + r*NE_ + j1)*2 + 1] : 0.f;
    }
    sched_fence();
    // receiver features (same for whole group -> broadcast A-fragment)
    v2f a_recv = *(const v2f*)&lds[OFF_XS + r*4 + ko];

#pragma unroll 1
    for (int jt = 0; jt < 4; ++jt) {           // 4 edge-tiles of 16 (63 real + 1 dummy)
      int j = jt*16 + l15;
      int s = (j < r) ? j : j + 1;
      if (j == NE_) s = r;                     // dummy edge, weight already 0
      v2f a_send = *(const v2f*)&lds[OFF_XS + s*4 + ko];

      sched_fence();                           // WAR fence vs. previous tile's hb reads
      // fc1: h(16x64) = relu(pre(16x8) @ W1^T + b1), staged to hb
#pragma unroll
      for (int ct = 0; ct < 4; ++ct) {
        v8f c = {};
        c = wmma4(a_recv, b1f[ct][0], c);      // K = 0..3  (recv features)
        c = wmma4(a_send, b1f[ct][1], c);      // K = 4..7  (send features)
#pragma unroll
        for (int v = 0; v < 8; ++v) {
          float hval = c[v] + b1v[ct];
          hval = hval > 0.f ? hval : 0.f;
          hb[(v + (hi ? 8 : 0))*HB_STR + 16*ct + l15] = hval;
        }
      }
      sched_fence();                           // RAW fence: hb stores before A reloads
      // reload h as fp32 A-fragments (16 K-chunks of 4)
      v2f af[16];
#pragma unroll
      for (int kc = 0; kc < 16; ++kc)
        af[kc] = *(const v2f*)&hb[l15*HB_STR + 4*kc + ko];

      // fc2: m(16x64) = relu(h @ W2^T + b2); weighted edge-sum -> agg[r]
#pragma unroll
      for (int ct = 0; ct < 4; ++ct) {
        v8f c = {};
#pragma unroll
        for (int kc = 0; kc < 16; ++kc) {
          v2f bv = *(const v2f*)&lds[OFF_W2 + (16*ct + l15)*W2_STR + 4*kc + ko];
          c = wmma4(af[kc], bv, c);
        }
        float partial = 0.f;
#pragma unroll
        for (int v = 0; v < 8; ++v) {
          float m = c[v] + b2v[ct];
          m = m > 0.f ? m : 0.f;
          partial += m * lds[OFF_WB + wave*64 + jt*16 + v + (hi ? 8 : 0)];
        }
        // lanes l and l+16 hit the same feature -> 2-way LDS fp32 atomic
        atomicAdd(&lds[OFF_AGG + r*AGG_STR + 16*ct + l15], partial);
      }
    }
  }
  __syncthreads();

  // ---------------- node phase: out1 (aug 68 -> 64) ----------------
#pragma unroll
  for (int jj = 0; jj < 2; ++jj) {
    int job = wave + 8*jj;                     // 16 (mt,ct) jobs over 8 waves
    int mt = job >> 2, ct = job & 3;
    int n = 16*mt + l15;
    v8f c = {};
    { // kc = 0: x part of aug
      v2f a  = *(const v2f*)&lds[OFF_XS + n*4 + ko];
      v2f bv = *(const v2f*)&lds[OFF_O1 + (16*ct + l15)*O1_STR + ko];
      c = wmma4(a, bv, c);
    }
#pragma unroll
    for (int kc = 1; kc < 17; ++kc) {          // agg part of aug
      v2f a  = *(const v2f*)&lds[OFF_AGG + n*AGG_STR + 4*(kc-1) + ko];
      v2f bv = *(const v2f*)&lds[OFF_O1 + (16*ct + l15)*O1_STR + 4*kc + ko];
      c = wmma4(a, bv, c);
    }
    float bb = lds[OFF_O1B + 16*ct + l15];
#pragma unroll
    for (int v = 0; v < 8; ++v) {
      float hv = c[v] + bb;
      hv = hv > 0.f ? hv : 0.f;
      lds[OFF_HB + (16*mt + v + (hi ? 8 : 0))*HB_STR + 16*ct + l15] = hv;  // h1
    }
  }
  __syncthreads();

  // ---------------- node phase: out2 (64 -> 64) ----------------
#pragma unroll
  for (int jj = 0; jj < 2; ++jj) {
    int job = wave + 8*jj;
    int mt = job >> 2, ct = job & 3;
    int n = 16*mt + l15;
    v8f c = {};
#pragma unroll
    for (int kc = 0; kc < 16; ++kc) {
      v2f a  = *(const v2f*)&lds[OFF_HB + n*HB_STR + 4*kc + ko];           // h1
      v2f bv = *(const v2f*)&lds[OFF_O2 + (16*ct + l15)*O2_STR + 4*kc + ko];
      c = wmma4(a, bv, c);
    }
    float bb = lds[OFF_O2B + 16*ct + l15];
#pragma unroll
    for (int v = 0; v < 8; ++v) {
      float hv = c[v] + bb;
      hv = hv > 0.f ? hv : 0.f;
      lds[OFF_HB + 64*HB_STR + (16*mt + v + (hi ? 8 : 0))*HB_STR + 16*ct + l15] = hv; // h2
    }
  }
  __syncthreads();

  // ---------------- out3 (64 -> 4) + residual + store ----------------
  {
    int n = tid >> 2, d = tid & 3;
    float acc = lds[OFF_O3B + d];
    const float* h2 = &lds[OFF_HB + 64*HB_STR + n*HB_STR];
    const float* w3 = &lds[OFF_O3W + d*64];
#pragma unroll 8
    for (int k = 0; k < 64; ++k) acc += h2[k] * w3[k];
    // pred_all = x_tp1[:, :T-1], swapped back to (B, N, T-1, D)
    out[((b*N_ + n)*(T_-1) + t)*D_ + d] = lds[OFF_XS + n*4 + d] + acc;
  }
}

extern "C" void kernel_launch(void* const* d_in, const int* in_sizes, int n_in,
                              void* d_out, int out_size, void* d_ws, size_t ws_size,
                              hipStream_t stream) {
  (void)in_sizes; (void)n_in; (void)d_ws; (void)ws_size; (void)out_size;
  const float* inputs   = (const float*)d_in[0];
  const float* rel_type = (const float*)d_in[1];
  // d_in[2] rel_rec / d_in[3] rel_send: one-hot gather matrices, indices computed analytically
  const float* fc1w = (const float*)d_in[4];
  const float* fc1b = (const float*)d_in[5];
  const float* fc2w = (const float*)d_in[6];
  const float* fc2b = (const float*)d_in[7];
  const float* o1w  = (const float*)d_in[8];
  const float* o1b  = (const float*)d_in[9];
  const float* o2w  = (const float*)d_in[10];
  const float* o2b  = (const float*)d_in[11];
  const float* o3w  = (const float*)d_in[12];
  const float* o3b  = (const float*)d_in[13];
  // d_in[14] pred_steps == 1 (asserted by reference)

  dim3 grid(T_ - 1, B_);   // 49 x 8 = 392 workgroups
  nri_decoder_kernel<<<grid, 256, LDS_FLOATS * sizeof(float), stream>>>(
      inputs, rel_type, fc1w, fc1b, fc2w, fc2b,
      o1w, o1b, o2w, o2b, o3w, o3b, (float*)d_out);
}